// PredictorFC_13434657702491
// MI455X (gfx1250) — compile-verified
//
#include <hip/hip_runtime.h>

typedef float v2f __attribute__((ext_vector_type(2)));
typedef float v8f __attribute__((ext_vector_type(8)));

#define BS     8192
#define HIDDEN 64

// ---------------------------------------------------------------------------
// Stage 1: Y = X @ W   (8192x64 @ 64x2 -> 8192x2), tiny, memory-trivial.
// ---------------------------------------------------------------------------
__global__ void fc_proj_kernel(const float* __restrict__ X,
                               const float* __restrict__ W,
                               float* __restrict__ Y) {
    int t = blockIdx.x * blockDim.x + threadIdx.x;   // 16384 threads
    int row = t >> 1;
    int c   = t & 1;
    const float* xr = X + (size_t)row * HIDDEN;
    float acc = 0.0f;
#pragma unroll
    for (int h = 0; h < HIDDEN; ++h)
        acc = fmaf(xr[h], W[h * 2 + c], acc);
    Y[row * 2 + c] = acc;
}

// ---------------------------------------------------------------------------
// Stage 2: out[i,:] = b + sum_{j in seg(i)} A[i,j] * Y[j,:]
// One wave32 per 16-row tile; v_wmma_f32_16x16x4_f32, A-block (16xK) x
// Yext (Kx16, cols>=2 zero). Y rows are broadcast-loaded as float2 in all
// lanes and component-selected branch-free, so the WMMA loops contain no
// EXEC-divergent control flow. B needs no masking: wherever a k-column is
// invalid for a row, the A element is zeroed, so the product vanishes.
// ---------------------------------------------------------------------------
__global__ void blockdiag_agg_kernel(const float* __restrict__ A,    // 8192 x 8192
                                     const int*   __restrict__ seg,  // 8192 (sorted)
                                     const float* __restrict__ Y,    // 8192 x 2
                                     const float* __restrict__ bias, // 2
                                     float* __restrict__ out) {      // 8192 x 2
    const int lane    = threadIdx.x & 31;
    const int tile    = (blockIdx.x * blockDim.x + threadIdx.x) >> 5; // 0..511
    const int rowbase = tile * 16;
    const int mn      = lane & 15;   // A/C row index == B/C column index
    const int hi      = lane >> 4;   // lane-half: K offset for A/B, M+8 for C/D
    const int row     = rowbase + mn;

    // Segment span of this tile (identical in all lanes; force SGPR).
    const int v0  = seg[rowbase];
    const int v15 = seg[rowbase + 15];
    int lo = 0, up = BS;
    while (lo < up) { int mid = (lo + up) >> 1; if (seg[mid] <  v0)  lo = mid + 1; else up = mid; }
    int jstart = lo;
    lo = 0; up = BS;
    while (lo < up) { int mid = (lo + up) >> 1; if (seg[mid] <= v15) lo = mid + 1; else up = mid; }
    int jend = lo;
    jstart = __builtin_amdgcn_readfirstlane(jstart);
    jend   = __builtin_amdgcn_readfirstlane(jend);
    const bool tile_uniform =
        (__builtin_amdgcn_readfirstlane((int)(v0 == v15)) != 0);

    const int    segRow = seg[row];
    const float* arow   = A + (size_t)row * BS;

    const bool is0 = (mn == 0);
    const bool is1 = (mn == 1);

    v8f acc = {0.f, 0.f, 0.f, 0.f, 0.f, 0.f, 0.f, 0.f};

    int j = jstart;
    if (tile_uniform) {
        // Interior tile: whole K range valid for every row -> no masking.
        // Unrolled x2: both load groups issue before the two WMMAs.
        for (; j + 8 <= jend; j += 8) {
            const int k0 = j + 2 * hi;      // ISA A-layout: k = v + 2*(lane>>4)
            const int k4 = k0 + 4;
            v2f a0, a1;
            a0.x = arow[k0];     a0.y = arow[k0 + 1];
            a1.x = arow[k4];     a1.y = arow[k4 + 1];
            const v2f y00 = *(const v2f*)(Y + (size_t)(k0    ) * 2);
            const v2f y01 = *(const v2f*)(Y + (size_t)(k0 + 1) * 2);
            const v2f y10 = *(const v2f*)(Y + (size_t)(k4    ) * 2);
            const v2f y11 = *(const v2f*)(Y + (size_t)(k4 + 1) * 2);
            __builtin_prefetch(arow + j + 64, 0, 3);   // near-cache stream-ahead
            v2f b0, b1;
            b0.x = is0 ? y00.x : (is1 ? y00.y : 0.0f);
            b0.y = is0 ? y01.x : (is1 ? y01.y : 0.0f);
            b1.x = is0 ? y10.x : (is1 ? y10.y : 0.0f);
            b1.y = is0 ? y11.x : (is1 ? y11.y : 0.0f);
            acc = __builtin_amdgcn_wmma_f32_16x16x4_f32(
                false, a0, false, b0, (short)0, acc, false, false);
            acc = __builtin_amdgcn_wmma_f32_16x16x4_f32(
                false, a1, false, b1, (short)0, acc, false, false);
        }
        for (; j + 4 <= jend; j += 4) {
            const int k0 = j + 2 * hi;
            v2f a, bf;
            a.x = arow[k0];
            a.y = arow[k0 + 1];
            const v2f y0 = *(const v2f*)(Y + (size_t)(k0    ) * 2);
            const v2f y1 = *(const v2f*)(Y + (size_t)(k0 + 1) * 2);
            bf.x = is0 ? y0.x : (is1 ? y0.y : 0.0f);
            bf.y = is0 ? y1.x : (is1 ? y1.y : 0.0f);
            acc = __builtin_amdgcn_wmma_f32_16x16x4_f32(
                false, a, false, bf, (short)0, acc, false, false);
        }
    }
    // Masked loop: K tail and/or tiles straddling a segment boundary.
    // Addresses clamped (branch-free); only A is predicated.
    const int last = jend - 1;
    for (; j < jend; j += 4) {
        const int  k0  = j + 2 * hi;
        const int  k1  = k0 + 1;
        const bool c0  = (k0 < jend);
        const bool c1  = (k1 < jend);
        const int  kc0 = c0 ? k0 : last;
        const int  kc1 = c1 ? k1 : last;
        const bool p0  = c0 && (seg[kc0] == segRow);  // block-diagonal mask on A
        const bool p1  = c1 && (seg[kc1] == segRow);
        const float av0 = arow[kc0];
        const float av1 = arow[kc1];
        const v2f   y0  = *(const v2f*)(Y + (size_t)kc0 * 2);
        const v2f   y1  = *(const v2f*)(Y + (size_t)kc1 * 2);
        v2f a, bf;
        a.x  = p0 ? av0 : 0.0f;
        a.y  = p1 ? av1 : 0.0f;
        bf.x = is0 ? y0.x : (is1 ? y0.y : 0.0f);
        bf.y = is0 ? y1.x : (is1 ? y1.y : 0.0f);
        acc = __builtin_amdgcn_wmma_f32_16x16x4_f32(
            false, a, false, bf, (short)0, acc, false, false);
    }

    // D layout: lanes 0-15 -> M = r, lanes 16-31 -> M = r + 8; N = lane & 15.
    if (mn < 2) {
        const float bv = bias[mn];
#pragma unroll
        for (int r = 0; r < 8; ++r) {
            const int mrow = rowbase + r + 8 * hi;
            out[(size_t)mrow * 2 + mn] = acc[r] + bv;
        }
    }
}

// ---------------------------------------------------------------------------
extern "C" void kernel_launch(void* const* d_in, const int* in_sizes, int n_in,
                              void* d_out, int out_size, void* d_ws, size_t ws_size,
                              hipStream_t stream) {
    const float* X   = (const float*)d_in[0];   // coded_tracks [8192,64]
    const float* A   = (const float*)d_in[1];   // attentions   [8192,8192]
    const int*   seg = (const int*)  d_in[2];   // seg_ids      [8192] sorted
    const float* W   = (const float*)d_in[3];   // [64,2]
    const float* b   = (const float*)d_in[4];   // [2]
    float*       out = (float*)d_out;           // [8192,2]
    float*       Y   = (float*)d_ws;            // [8192,2] scratch (64 KB)

    // Stage 1: 16384 threads = 64 blocks x 256
    fc_proj_kernel<<<64, 256, 0, stream>>>(X, W, Y);
    // Stage 2: 512 tiles x 1 wave32 = 16384 threads = 64 blocks x 256 (8 waves)
    blockdiag_agg_kernel<<<64, 256, 0, stream>>>(A, seg, Y, b, out);
}